// LS_44160853737780
// MI455X (gfx1250) — compile-verified
//
#include <hip/hip_runtime.h>
#include <cstddef>
#include <cstdint>

// Reference: img/light [B, 96, 256, 256] fp32 -> out [B, 3, 256, 256] fp32.
// Memory-bound (~1 FLOP/byte, 204 MB traffic): maximize bytes in flight,
// stream with non-temporal policy (dataset > 192 MB L2).
#define PLANE 65536         // H*W
#define NCH   96
#define NF    32            // lights
#define CSTRB (PLANE * 4)   // bytes per channel plane = 256 KB

typedef float v4f __attribute__((ext_vector_type(4)));

// gfx1250 async global->LDS copy (ASYNCcnt-tracked, no VGPR destination).
// GVS mode: mem = SADDR(sgpr64) + VADDR(vgpr32), LDS addr from VGPR.
// th:TH_LOAD_NT -> non-temporal: read-once stream, don't pollute WGP$/L2.
#define ASYNC_LOAD_B128(ldsaddr, voffset, sbase)                              \
  asm volatile("global_load_async_to_lds_b128 %0, %1, %2 th:TH_LOAD_NT"       \
               :: "v"(ldsaddr), "v"(voffset), "s"(sbase) : "memory")

#define WAITA(n)  asm volatile("s_wait_asynccnt " #n ::: "memory")
#define WAITDS()  asm volatile("s_wait_dscnt 0" ::: "memory")

// Stage one light into LDS slot s, channel-0 byte offset given explicitly.
// LDS layout: [slot 0..3][buf 0..5][tid]*16B ; 4096 B per buf, 24 KB per slot.
#define ISSUE_AT(c0off, s) do {                                               \
    const unsigned v0 = (c0off);                                              \
    const unsigned v1 = (c0off) + (unsigned)CSTRB;                            \
    const unsigned v2 = (c0off) + (unsigned)(2 * CSTRB);                      \
    const unsigned lb = ldst + (unsigned)((s) * 6 * 4096);                    \
    ASYNC_LOAD_B128(lb +      0u, v0, imgb);                                  \
    ASYNC_LOAD_B128(lb +   4096u, v1, imgb);                                  \
    ASYNC_LOAD_B128(lb +   8192u, v2, imgb);                                  \
    ASYNC_LOAD_B128(lb +  12288u, v0, lgtb);                                  \
    ASYNC_LOAD_B128(lb +  16384u, v1, lgtb);                                  \
    ASYNC_LOAD_B128(lb +  20480u, v2, lgtb);                                  \
  } while (0)

// Consume LDS slot s: 6 x ds_load_b128, accumulate LtL (sym 6) and LtI (3)
// for this thread's 4 pixels.
#define CONSUME(s) do {                                                       \
    const float4 ir = smem[((s) * 6 + 0) * 256 + t];                          \
    const float4 ig = smem[((s) * 6 + 1) * 256 + t];                          \
    const float4 ib = smem[((s) * 6 + 2) * 256 + t];                          \
    const float4 lr = smem[((s) * 6 + 3) * 256 + t];                          \
    const float4 lg = smem[((s) * 6 + 4) * 256 + t];                          \
    const float4 lb = smem[((s) * 6 + 5) * 256 + t];                          \
    const float IR[4] = {ir.x, ir.y, ir.z, ir.w};                             \
    const float IG[4] = {ig.x, ig.y, ig.z, ig.w};                             \
    const float IB[4] = {ib.x, ib.y, ib.z, ib.w};                             \
    const float LR[4] = {lr.x, lr.y, lr.z, lr.w};                             \
    const float LG[4] = {lg.x, lg.y, lg.z, lg.w};                             \
    const float LB[4] = {lb.x, lb.y, lb.z, lb.w};                             \
    _Pragma("unroll")                                                         \
    for (int k = 0; k < 4; ++k) {                                             \
      const float Iv = __builtin_amdgcn_sqrtf(                                \
          IR[k]*IR[k] + IG[k]*IG[k] + IB[k]*IB[k]);                           \
      t0[k]  += LR[k] * Iv;                                                   \
      t1[k]  += LG[k] * Iv;                                                   \
      t2[k]  += LB[k] * Iv;                                                   \
      s00[k] += LR[k] * LR[k];                                                \
      s01[k] += LR[k] * LG[k];                                                \
      s02[k] += LR[k] * LB[k];                                                \
      s11[k] += LG[k] * LG[k];                                                \
      s12[k] += LG[k] * LB[k];                                                \
      s22[k] += LB[k] * LB[k];                                                \
    }                                                                         \
  } while (0)

__global__ __launch_bounds__(256) void photometric_normals(
    const float* __restrict__ img, const float* __restrict__ light,
    float* __restrict__ out) {
  const int t   = (int)threadIdx.x;
  // 64 blocks per plane -> batch index derives from blockIdx only (uniform,
  // so the async SADDR base provably lives in an SGPR pair).
  const int bb  = (int)(blockIdx.x >> 6);
  const int blk = (int)(blockIdx.x & 63);
  const int qq  = (blk << 8) + t;              // float4-quad index in plane

  const float* imgb = img   + (size_t)bb * NCH * PLANE;
  const float* lgtb = light + (size_t)bb * NCH * PLANE;

  __shared__ float4 smem[4 * 6 * 256];         // 96 KB, 4-slot ring buffer
  const unsigned lds0 = (unsigned)(size_t)(&smem[0]);   // low 32b = LDS offset
  const unsigned ldst = lds0 + (unsigned)t * 16u;
  const unsigned vq   = (unsigned)qq * 16u;    // byte offset of quad in plane

  float s00[4], s01[4], s02[4], s11[4], s12[4], s22[4];
  float t0[4], t1[4], t2[4];
#pragma unroll
  for (int k = 0; k < 4; ++k) {
    s00[k] = s01[k] = s02[k] = s11[k] = s12[k] = s22[k] = 0.f;
    t0[k] = t1[k] = t2[k] = 0.f;
  }

  // Prime: lights 0..3 into slots 0..3 (24 async ops in flight).
  ISSUE_AT(vq + 0u * 3u * CSTRB, 0);
  ISSUE_AT(vq + 1u * 3u * CSTRB, 1);
  ISSUE_AT(vq + 2u * 3u * CSTRB, 2);
  ISSUE_AT(vq + 3u * 3u * CSTRB, 3);

  // Rolled steady-state loop: one running address register (vbase) instead
  // of ~100 hoisted VGPRs; pipeline stays 3 lights (18 async ops) deep.
  unsigned vbase = vq + 12u * (unsigned)CSTRB;   // light 4, channel 0
#pragma clang loop unroll(disable)
  for (int i = 0; i < 7; ++i) {                  // lights 4i .. 4i+3
    WAITA(18); CONSUME(0); WAITDS(); ISSUE_AT(vbase + 0u * 3u * CSTRB, 0);
    WAITA(18); CONSUME(1); WAITDS(); ISSUE_AT(vbase + 1u * 3u * CSTRB, 1);
    WAITA(18); CONSUME(2); WAITDS(); ISSUE_AT(vbase + 2u * 3u * CSTRB, 2);
    WAITA(18); CONSUME(3); WAITDS(); ISSUE_AT(vbase + 3u * 3u * CSTRB, 3);
    vbase += 12u * (unsigned)CSTRB;
  }

  // Drain: lights 28..31, descending outstanding counts.
  WAITA(18); CONSUME(0);
  WAITA(12); CONSUME(1);
  WAITA(6);  CONSUME(2);
  WAITA(0);  CONSUME(3);

  // Per-pixel 3x3 symmetric inverse (adjugate) and normalize.
  float ox[4], oy[4], oz[4];
#pragma unroll
  for (int k = 0; k < 4; ++k) {
    const float a = s00[k], b = s01[k], c = s02[k];
    const float d = s11[k], e = s12[k], g = s22[k];
    const float c00 = d * g - e * e;
    const float c01 = c * e - b * g;
    const float c02 = b * e - c * d;
    const float c11 = a * g - c * c;
    const float c12 = c * b - a * e;
    const float c22 = a * d - b * b;
    const float det = a * c00 + b * c01 + c * c02;
    const float id  = __builtin_amdgcn_rcpf(det);
    const float nx = (c00 * t0[k] + c01 * t1[k] + c02 * t2[k]) * id;
    const float ny = (c01 * t0[k] + c11 * t1[k] + c12 * t2[k]) * id;
    const float nz = (c02 * t0[k] + c12 * t1[k] + c22 * t2[k]) * id;
    const float rn = __builtin_amdgcn_rcpf(
        __builtin_amdgcn_sqrtf(nx*nx + ny*ny + nz*nz) + 1e-10f);
    ox[k] = nx * rn;
    oy[k] = ny * rn;
    oz[k] = nz * rn;
  }

  // Output written once, never re-read: non-temporal b128 stores.
  float* outb = out + (size_t)bb * 3 * PLANE;
  const v4f v0 = {ox[0], ox[1], ox[2], ox[3]};
  const v4f v1 = {oy[0], oy[1], oy[2], oy[3]};
  const v4f v2 = {oz[0], oz[1], oz[2], oz[3]};
  __builtin_nontemporal_store(v0, (v4f*)(outb + 0 * PLANE + (size_t)qq * 4));
  __builtin_nontemporal_store(v1, (v4f*)(outb + 1 * PLANE + (size_t)qq * 4));
  __builtin_nontemporal_store(v2, (v4f*)(outb + 2 * PLANE + (size_t)qq * 4));
}

extern "C" void kernel_launch(void* const* d_in, const int* in_sizes, int n_in,
                              void* d_out, int out_size, void* d_ws, size_t ws_size,
                              hipStream_t stream) {
  (void)n_in; (void)out_size; (void)d_ws; (void)ws_size;
  const float* img   = (const float*)d_in[0];
  const float* light = (const float*)d_in[1];
  float* out = (float*)d_out;

  const int B = in_sizes[0] / (NCH * PLANE);           // 4 for the reference
  const int blocks = B * (PLANE / (4 * 256));          // 64 blocks per plane
  photometric_normals<<<blocks, 256, 0, stream>>>(img, light, out);
}